// GATConvLayer_21492016349946
// MI455X (gfx1250) — compile-verified
//
#include <hip/hip_runtime.h>

#define N_NODES 50000
#define N_EDGES 800000
#define DIM     128
#define NH      4
#define HD      32

typedef __attribute__((ext_vector_type(2))) float v2f;
typedef __attribute__((ext_vector_type(8))) float v8f;

// ---------------------------------------------------------------------------
// Kernel 1: collapse W_dst against attn_dst:  w_eff[k][h], c_dst[h]
// (x_dst is only ever consumed via sum_d x_dst*attn_dst, so never build it)
// ---------------------------------------------------------------------------
__global__ void k_prep(const float* __restrict__ W_dst,
                       const float* __restrict__ b_dst,
                       const float* __restrict__ attn_dst,
                       float* __restrict__ w_eff,
                       float* __restrict__ c_dst) {
    const int k = threadIdx.x;  // 0..127
    for (int h = 0; h < NH; ++h) {
        float s = 0.f;
        #pragma unroll 8
        for (int d = 0; d < HD; ++d)
            s += W_dst[k * DIM + h * HD + d] * attn_dst[h * HD + d];
        w_eff[k * NH + h] = s;
    }
    if (k < NH) {
        float s = 0.f;
        for (int d = 0; d < HD; ++d)
            s += b_dst[k * HD + d] * attn_dst[k * HD + d];
        c_dst[k] = s;
    }
}

// ---------------------------------------------------------------------------
// Kernel 2: Xsrc = x @ W_src + b_src  via V_WMMA_F32_16X16X4_F32 (wave32).
// One 16x16 output tile per wave; 8 waves/block cover all 128 columns of a
// 16-row block. K=128 stepped by 4 -> 32 WMMA ops per tile.
// A layout (16x4 f32): lane&15 = M row; (lane>>4)*2 = K pair base; v0=K, v1=K+1
// B layout (4x16 f32): lane&15 = N col; same K-pair split across lane halves
// C/D layout: VGPR i holds M=i (lanes 0-15) / M=8+i (lanes 16-31), N=lane&15
// ---------------------------------------------------------------------------
__global__ __launch_bounds__(256) void k_gemm(const float* __restrict__ x,
                                              const float* __restrict__ W,
                                              const float* __restrict__ bias,
                                              float* __restrict__ Y) {
    const int lane = threadIdx.x & 31;
    const int wave = threadIdx.x >> 5;            // 0..7 -> column tile
    const int row0 = blockIdx.x * 16;             // N_NODES % 16 == 0
    const int col  = wave * 16 + (lane & 15);
    const int kk   = (lane >> 4) * 2;             // 0 or 2
    const float* xrow = x + (size_t)(row0 + (lane & 15)) * DIM;

    v8f acc = {};
    #pragma unroll
    for (int k = 0; k < DIM; k += 4) {
        v2f a, b;
        a.x = xrow[k + kk];
        a.y = xrow[k + kk + 1];
        b.x = W[(size_t)(k + kk) * DIM + col];
        b.y = W[(size_t)(k + kk + 1) * DIM + col];
        acc = __builtin_amdgcn_wmma_f32_16x16x4_f32(
            false, a, false, b, (short)0, acc, false, false);
    }
    const float bb = bias[col];
    const int rbase = row0 + (lane >> 4) * 8;
    #pragma unroll
    for (int i = 0; i < 8; ++i)
        Y[(size_t)(rbase + i) * DIM + col] = acc[i] + bb;
}

// ---------------------------------------------------------------------------
// Kernel 3: per-node attention scores, one thread per (node, head).
//   s_src[n,h] = sum_d Xsrc[n, h*32+d] * attn_src[h,d]
//   s_dst[n,h] = sum_k x[n,k] * w_eff[k,h] + c_dst[h]
// Also zero-inits denom[n,h] for the upcoming segment softmax.
// ---------------------------------------------------------------------------
__global__ __launch_bounds__(256) void k_scores(const float* __restrict__ x,
                                                const float* __restrict__ Xsrc,
                                                const float* __restrict__ attn_src,
                                                const float* __restrict__ w_eff,
                                                const float* __restrict__ c_dst,
                                                float* __restrict__ s_src,
                                                float* __restrict__ s_dst,
                                                float* __restrict__ denom) {
    const int t = blockIdx.x * blockDim.x + threadIdx.x;
    if (t >= N_NODES * NH) return;
    const int n = t >> 2, h = t & 3;

    float ss = 0.f;
    const float* xr = Xsrc + (size_t)n * DIM + h * HD;
    #pragma unroll 8
    for (int d = 0; d < HD; ++d) ss += xr[d] * attn_src[h * HD + d];

    float sd = c_dst[h];
    const float* xi = x + (size_t)n * DIM;
    #pragma unroll 8
    for (int k = 0; k < DIM; ++k) sd += xi[k] * w_eff[k * NH + h];

    s_src[t] = ss;
    s_dst[t] = sd;
    denom[t] = 0.f;
}

// ---------------------------------------------------------------------------
// Kernel 4: zero the output accumulator (d_out is poisoned by the harness).
// ---------------------------------------------------------------------------
__global__ void k_zero(float* __restrict__ p, int n) {
    const int t = blockIdx.x * blockDim.x + threadIdx.x;
    if (t < n) p[t] = 0.f;
}

// ---------------------------------------------------------------------------
// Kernel 5: per-edge logits + exp + segment-sum denominator.
// One thread per (edge, head).
// ---------------------------------------------------------------------------
__global__ __launch_bounds__(256) void k_edge_logits(
        const long long* __restrict__ eidx,
        const float* __restrict__ edge_attr,
        const float* __restrict__ W_edge,
        const float* __restrict__ b_edge,
        const float* __restrict__ s_src,
        const float* __restrict__ s_dst,
        float* __restrict__ exp_a,
        float* __restrict__ denom) {
    const int t = blockIdx.x * blockDim.x + threadIdx.x;  // e*4 + h
    if (t >= N_EDGES * NH) return;
    const int e = t >> 2, h = t & 3;
    const int src = (int)eidx[e];
    const int dst = (int)eidx[N_EDGES + e];

    float a = s_src[src * NH + h] + s_dst[dst * NH + h];
    a = (a >= 0.f) ? a : 0.2f * a;                        // leaky_relu(0.2)
    a += edge_attr[e] * W_edge[h] + b_edge[h];            // edge bias
    a = fminf(fmaxf(a, -5.f), 5.f);                       // clip
    const float ex = expf(a);
    exp_a[t] = ex;
    atomicAdd(&denom[dst * NH + h], ex);
}

// ---------------------------------------------------------------------------
// Kernel 6: weighted message aggregation. One wave32 per edge; each lane
// handles 4 consecutive features (float4 gather of Xsrc row, 4 L2 atomics).
// Both Xsrc (25.6 MB) and out (25.6 MB) are L2-resident on MI455X (192 MB).
// ---------------------------------------------------------------------------
__global__ __launch_bounds__(256) void k_aggregate(
        const long long* __restrict__ eidx,
        const float* __restrict__ Xsrc,
        const float* __restrict__ exp_a,
        const float* __restrict__ denom,
        float* __restrict__ out) {
    const int lane = threadIdx.x & 31;
    const int e = blockIdx.x * 8 + (threadIdx.x >> 5);    // E % 8 == 0
    const int src = (int)eidx[e];
    const int dst = (int)eidx[N_EDGES + e];
    const int h = lane >> 3;                              // 8 lanes' features share a head (8 lanes * 4 feat = 32 = HD)

    const float attn = exp_a[e * NH + h] / (denom[dst * NH + h] + 1e-8f);
    const float4 v = *((const float4*)(Xsrc + (size_t)src * DIM) + lane);
    float* o = out + (size_t)dst * DIM + lane * 4;
    atomicAdd(o + 0, v.x * attn);
    atomicAdd(o + 1, v.y * attn);
    atomicAdd(o + 2, v.z * attn);
    atomicAdd(o + 3, v.w * attn);
}

// ---------------------------------------------------------------------------
// Kernel 7: exact GELU + residual + LayerNorm, one wave32 per node,
// 4 features per lane, wave shuffle reductions (wave32 -> xor 16..1).
// ---------------------------------------------------------------------------
__global__ __launch_bounds__(256) void k_finalize(const float* __restrict__ x,
                                                  const float* __restrict__ gamma,
                                                  const float* __restrict__ beta,
                                                  float* __restrict__ out) {
    const int lane = threadIdx.x & 31;
    const int n = blockIdx.x * 8 + (threadIdx.x >> 5);    // N % 8 == 0
    float* o = out + (size_t)n * DIM + lane * 4;
    const float* xr = x + (size_t)n * DIM + lane * 4;

    float hv[4];
    float s = 0.f, s2 = 0.f;
    #pragma unroll
    for (int j = 0; j < 4; ++j) {
        const float v = o[j];
        const float g = 0.5f * v * (1.f + erff(v * 0.70710678118654752f));
        hv[j] = g + xr[j];
        s  += hv[j];
        s2 += hv[j] * hv[j];
    }
    #pragma unroll
    for (int off = 16; off >= 1; off >>= 1) {
        s  += __shfl_xor(s,  off, 32);
        s2 += __shfl_xor(s2, off, 32);
    }
    const float mu  = s * (1.f / 128.f);
    const float var = s2 * (1.f / 128.f) - mu * mu;
    const float inv = rsqrtf(var + 1e-5f);
    #pragma unroll
    for (int j = 0; j < 4; ++j)
        o[j] = (hv[j] - mu) * inv * gamma[lane * 4 + j] + beta[lane * 4 + j];
}

// ---------------------------------------------------------------------------
extern "C" void kernel_launch(void* const* d_in, const int* in_sizes, int n_in,
                              void* d_out, int out_size, void* d_ws, size_t ws_size,
                              hipStream_t stream) {
    const float*     x         = (const float*)d_in[0];
    const long long* eidx      = (const long long*)d_in[1];   // int64 [2,E]
    const float*     edge_attr = (const float*)d_in[2];
    const float*     W_src     = (const float*)d_in[3];
    const float*     b_src     = (const float*)d_in[4];
    const float*     W_dst     = (const float*)d_in[5];
    const float*     b_dst     = (const float*)d_in[6];
    const float*     W_edge    = (const float*)d_in[7];
    const float*     b_edge    = (const float*)d_in[8];
    const float*     attn_src  = (const float*)d_in[9];
    const float*     attn_dst  = (const float*)d_in[10];
    const float*     gamma     = (const float*)d_in[11];
    const float*     beta      = (const float*)d_in[12];
    float* out = (float*)d_out;

    // workspace layout (floats), ~41 MB total
    float* Xsrc  = (float*)d_ws;                              // N*128
    float* s_src = Xsrc  + (size_t)N_NODES * DIM;             // N*4
    float* s_dst = s_src + (size_t)N_NODES * NH;              // N*4
    float* denom = s_dst + (size_t)N_NODES * NH;              // N*4
    float* w_eff = denom + (size_t)N_NODES * NH;              // 128*4
    float* c_dst = w_eff + DIM * NH;                          // 4 (padded)
    float* exp_a = c_dst + 64;                                // E*4

    k_prep<<<1, 128, 0, stream>>>(W_dst, b_dst, attn_dst, w_eff, c_dst);
    k_gemm<<<N_NODES / 16, 256, 0, stream>>>(x, W_src, b_src, Xsrc);
    k_scores<<<(N_NODES * NH + 255) / 256, 256, 0, stream>>>(
        x, Xsrc, attn_src, w_eff, c_dst, s_src, s_dst, denom);
    k_zero<<<(N_NODES * DIM) / 256, 256, 0, stream>>>(out, N_NODES * DIM);
    k_edge_logits<<<(N_EDGES * NH + 255) / 256, 256, 0, stream>>>(
        eidx, edge_attr, W_edge, b_edge, s_src, s_dst, exp_a, denom);
    k_aggregate<<<N_EDGES / 8, 256, 0, stream>>>(eidx, Xsrc, exp_a, denom, out);
    k_finalize<<<N_NODES / 8, 256, 0, stream>>>(x, gamma, beta, out);
}